// GCN_25555055411820
// MI455X (gfx1250) — compile-verified
//
#include <hip/hip_runtime.h>

typedef __attribute__((ext_vector_type(2))) float v2f;
typedef __attribute__((ext_vector_type(8))) float v8f;

__device__ __forceinline__ void atomic_add_f32(float* p, float v) {
  // Relaxed agent-scope fp atomic -> global_atomic_add_f32 (no CAS loop)
  (void)__hip_atomic_fetch_add(p, v, __ATOMIC_RELAXED, __HIP_MEMORY_SCOPE_AGENT);
}

// ---------------- elementwise helpers ----------------

__global__ void gcn_zero_kernel(float* __restrict__ p, int n) {
  int i = blockIdx.x * blockDim.x + threadIdx.x;
  if (i < n) p[i] = 0.0f;
}

__global__ void gcn_degree_kernel(const int* __restrict__ src, const int* __restrict__ dst,
                                  float* __restrict__ deg_out, float* __restrict__ deg_in,
                                  int E) {
  int e = blockIdx.x * blockDim.x + threadIdx.x;
  if (e < E) {
    atomic_add_f32(&deg_out[src[e]], 1.0f);
    atomic_add_f32(&deg_in[dst[e]], 1.0f);
  }
}

__global__ void gcn_norm_kernel(float* __restrict__ deg, int n) {
  int i = blockIdx.x * blockDim.x + threadIdx.x;
  if (i < n) deg[i] = rsqrtf(fmaxf(deg[i], 1.0f));
}

// h[i,f] = relu(h[i,f] * norm_dst[i] + bias[f]); in place, n = nrows*H
__global__ void gcn_post_kernel(float* __restrict__ h, const float* __restrict__ norm_dst,
                                const float* __restrict__ bias, int n, int H) {
  int idx = blockIdx.x * blockDim.x + threadIdx.x;
  if (idx >= n) return;
  int i = idx / H;
  int f = idx - i * H;
  h[idx] = fmaxf(fmaf(h[idx], norm_dst[i], bias[f]), 0.0f);
}

// ---------------- WMMA GEMM: out[N, wcols] = (A[N, K] * norm[:,None]) @ W[K, wcols] ----------------
// One wave32 per 16-row tile. fp32 WMMA 16x16x4; wcols <= 32 covered by 2 N-tiles.
// W is staged once per block into a zero-padded 32x32 LDS tile so the K-loop has
// no predicated loads (branch-free: 1 global b64 + 4 ds loads + 2 WMMA per step).
__global__ void gcn_gemm_wmma_kernel(const float* __restrict__ A, int lda, int K,
                                     const float* __restrict__ W, int wcols,
                                     const float* __restrict__ norm,
                                     float* __restrict__ out, int ldo, int ntiles) {
  __shared__ float Ws[32 * 32];  // [k][n], stride 32, zero padded

  // cooperative zero-padded fill of W (K x wcols, K<=32, wcols<=32)
  for (int i = threadIdx.x; i < 32 * 32; i += blockDim.x) {
    const int k = i >> 5;
    const int n = i & 31;
    Ws[i] = (k < K && n < wcols) ? W[(size_t)k * wcols + n] : 0.0f;
  }
  __syncthreads();

  const int wave = (blockIdx.x * blockDim.x + threadIdx.x) >> 5;  // wave-uniform tile index
  const int lane = threadIdx.x & 31;
  if (wave >= ntiles) return;  // uniform branch: EXEC stays all-1s

  const int m = lane & 15;     // row within tile (also C/D column)
  const int half = lane >> 4;  // K-pair selector

  const int row = wave * 16 + m;
  const float nrm = norm[row];
  const float* arow = A + (size_t)row * lda;

  v8f c0 = {};  // output cols 0..15
  v8f c1 = {};  // output cols 16..31 (zero-padded past wcols)

  for (int k0 = 0; k0 < K; k0 += 4) {
    const int k = k0 + 2 * half;  // this lane provides K = k, k+1
    const float2 av = *(const float2*)(arow + k);
    v2f a;
    a.x = av.x * nrm;
    a.y = av.y * nrm;

    v2f b0, b1;
    b0.x = Ws[k * 32 + m];
    b0.y = Ws[(k + 1) * 32 + m];
    b1.x = Ws[k * 32 + 16 + m];
    b1.y = Ws[(k + 1) * 32 + 16 + m];

    c0 = __builtin_amdgcn_wmma_f32_16x16x4_f32(false, a, false, b0, (short)0, c0, false, false);
    c1 = __builtin_amdgcn_wmma_f32_16x16x4_f32(false, a, false, b1, (short)0, c1, false, false);
  }

  // C/D layout: VGPR v, lane -> row (v + 8*half), col m
  for (int v = 0; v < 8; ++v) {
    const int orow = wave * 16 + v + 8 * half;
    out[(size_t)orow * ldo + m] = c0[v];
    if (16 + m < wcols) out[(size_t)orow * ldo + 16 + m] = c1[v];
  }
}

// ---------------- edge scatter: agg[dst] += x[src], H=4*nch floats per node ----------------
// one thread per (edge, float4-chunk); nwork = E * nch
__global__ void gcn_scatter_kernel(const float* __restrict__ x, const int* __restrict__ src,
                                   const int* __restrict__ dst, float* __restrict__ agg,
                                   int nwork, int nch, int H) {
  int idx = blockIdx.x * blockDim.x + threadIdx.x;
  if (idx >= nwork) return;
  int e = idx / nch;
  int c = idx - e * nch;
  int s = src[e];
  int d = dst[e];
  const float4 v = *(const float4*)(x + (size_t)s * H + c * 4);
  float* ap = agg + (size_t)d * H + c * 4;
  atomic_add_f32(ap + 0, v.x);
  atomic_add_f32(ap + 1, v.y);
  atomic_add_f32(ap + 2, v.z);
  atomic_add_f32(ap + 3, v.w);
}

// ---------------- readout: out[r] = dot(h_flat[r*96 .. +96], Wd) + bd ----------------
__global__ void gcn_readout_kernel(const float* __restrict__ h, const float* __restrict__ Wd,
                                   const float* __restrict__ bd, float* __restrict__ out,
                                   int nq, int rowlen /* = 4*H floats */) {
  int r = blockIdx.x * blockDim.x + threadIdx.x;
  if (r >= nq) return;
  const float4* hp = (const float4*)(h + (size_t)r * rowlen);
  const float4* wp = (const float4*)Wd;
  const int nch = rowlen >> 2;
  float s = 0.0f;
  for (int j = 0; j < nch; ++j) {
    float4 hv = hp[j];
    float4 wv = wp[j];
    s = fmaf(hv.x, wv.x, s);
    s = fmaf(hv.y, wv.y, s);
    s = fmaf(hv.z, wv.z, s);
    s = fmaf(hv.w, wv.w, s);
  }
  out[r] = s + bd[0];
}

// ---------------- host launch ----------------

extern "C" void kernel_launch(void* const* d_in, const int* in_sizes, int n_in,
                              void* d_out, int out_size, void* d_ws, size_t ws_size,
                              hipStream_t stream) {
  const float* features = (const float*)d_in[0];
  const int*   src      = (const int*)d_in[1];
  const int*   dst      = (const int*)d_in[2];
  const float* W1       = (const float*)d_in[3];
  const float* b1       = (const float*)d_in[4];
  const float* W2       = (const float*)d_in[5];
  const float* b2       = (const float*)d_in[6];
  const float* Wd       = (const float*)d_in[7];
  const float* bd       = (const float*)d_in[8];
  float* out = (float*)d_out;

  const int H  = in_sizes[4];            // 24
  const int F  = in_sizes[3] / H;        // 32
  const int Nn = in_sizes[0] / F;        // 200000
  const int E  = in_sizes[1];            // 1600000

  // Workspace layout (floats): deg_out[Nn] | deg_in[Nn] | bufA[Nn*H] | bufX[Nn*H]
  float* deg_out = (float*)d_ws;
  float* deg_in  = deg_out + Nn;
  float* bufA    = deg_in + Nn;
  float* bufX    = bufA + (size_t)Nn * H;

  const int B = 256;
  const int NH = Nn * H;
  const int ntiles = Nn / 16;            // 200000 % 16 == 0
  const int wavesPerBlock = B / 32;
  const int gemmGrid = (ntiles + wavesPerBlock - 1) / wavesPerBlock;
  const int nch = H / 4;                 // float4 chunks per node row
  const int scatterWork = E * nch;
  const int nq = Nn / 4;

  // 1. zero degrees + first aggregation buffer (contiguous region)
  {
    int n = 2 * Nn + NH;
    gcn_zero_kernel<<<(n + B - 1) / B, B, 0, stream>>>(deg_out, n);
  }
  // 2. degrees
  gcn_degree_kernel<<<(E + B - 1) / B, B, 0, stream>>>(src, dst, deg_out, deg_in, E);
  // 3. norms in place (deg_out -> norm_src, deg_in -> norm_dst)
  gcn_norm_kernel<<<(2 * Nn + B - 1) / B, B, 0, stream>>>(deg_out, 2 * Nn);
  // 4. layer1 GEMM: bufX = (features * norm_src) @ W1
  gcn_gemm_wmma_kernel<<<gemmGrid, B, 0, stream>>>(features, F, F, W1, H, deg_out,
                                                   bufX, H, ntiles);
  // 5. scatter-add over edges
  gcn_scatter_kernel<<<(scatterWork + B - 1) / B, B, 0, stream>>>(bufX, src, dst, bufA,
                                                                  scatterWork, nch, H);
  // 6. h1 = relu(agg * norm_dst + b1) in place
  gcn_post_kernel<<<(NH + B - 1) / B, B, 0, stream>>>(bufA, deg_in, b1, NH, H);
  // 7. layer2 GEMM: bufX = (h1 * norm_src) @ W2
  gcn_gemm_wmma_kernel<<<gemmGrid, B, 0, stream>>>(bufA, H, H, W2, H, deg_out,
                                                   bufX, H, ntiles);
  // 8. re-zero aggregation buffer
  gcn_zero_kernel<<<(NH + B - 1) / B, B, 0, stream>>>(bufA, NH);
  // 9. scatter layer 2
  gcn_scatter_kernel<<<(scatterWork + B - 1) / B, B, 0, stream>>>(bufX, src, dst, bufA,
                                                                  scatterWork, nch, H);
  // 10. h2 = relu(agg * norm_dst + b2) in place
  gcn_post_kernel<<<(NH + B - 1) / B, B, 0, stream>>>(bufA, deg_in, b2, NH, H);
  // 11. readout: out[N/4,1] = reshape(h2) @ Wd + bd
  gcn_readout_kernel<<<(nq + B - 1) / B, B, 0, stream>>>(bufA, Wd, bd, out, nq, 4 * H);
}